// ROIPooler_20066087206999
// MI455X (gfx1250) — compile-verified
//
#include <hip/hip_runtime.h>

#define C_TOT 256
#define ROI 7
#define MAXB 8
#define CH_PER_BLK 8
#define TILE_H 32
#define TILE_W 32
#define TILE_WP 33   // padded row stride (floats) to break LDS bank aliasing

// CDNA5 async global->LDS load: vdst = LDS byte offset (VGPR), vaddr = 64-bit
// global address (VGPR pair), saddr = off. Tracked by ASYNCcnt.
__device__ __forceinline__ void async_ld_f32(const float* g, unsigned lds_off) {
  asm volatile("global_load_async_to_lds_b32 %0, %1, off"
               :: "v"(lds_off), "v"(g)
               : "memory");
}

__device__ __forceinline__ void wait_async_zero() {
  asm volatile("s_wait_asynccnt 0" ::: "memory");
}

struct RoiParams {
  const float* base;          // fts[level] + img*C*H*W
  int H, W, h0, w0, regH, regW;
  int hstart[ROI], hend[ROI], wstart[ROI], wend[ROI];
};

__global__ __launch_bounds__(256)
void ROIPooler_kernel(const float* __restrict__ ft0, const float* __restrict__ ft1,
                      const float* __restrict__ ft2, const float* __restrict__ ft3,
                      const float* __restrict__ rois, const int* __restrict__ roi_idx,
                      float* __restrict__ out) {
  __shared__ RoiParams P;
  __shared__ float tile[CH_PER_BLK][TILE_H][TILE_WP];  // ~33 KB, padded rows

  const int roi   = blockIdx.x;
  const int cbase = blockIdx.y * CH_PER_BLK;

  if (threadIdx.x == 0) {
    const float* fts[4] = {ft0, ft1, ft2, ft3};
    const int   HS[4] = {200, 100, 50, 25};
    const float SC[4] = {0.25f, 0.125f, 0.0625f, 0.03125f};

    const float y1 = rois[roi * 4 + 0];
    const float x1 = rois[roi * 4 + 1];
    const float y2 = rois[roi * 4 + 2];
    const float x2 = rois[roi * 4 + 3];

    // level assignment: floor(4 + log2(sqrt(h*w)/224)), clipped to [2,5]
    const float bs  = sqrtf((y2 - y1) * (x2 - x1));
    float lev = floorf(4.0f + log2f(bs * (1.0f / 224.0f)));
    lev = fminf(fmaxf(lev, 2.0f), 5.0f);
    const int li = (int)lev - 2;

    const int   H  = HS[li];
    const int   W  = HS[li];
    const float sc = SC[li];

    // round-half-up box edges in feature coords
    const int sh = (int)floorf(y1 * sc + 0.5f);
    const int sw = (int)floorf(x1 * sc + 0.5f);
    const int eh = (int)floorf(y2 * sc + 0.5f);
    const int ew = (int)floorf(x2 * sc + 0.5f);

    const float roih = (float)max(eh - sh + 1, 1);
    const float roiw = (float)max(ew - sw + 1, 1);
    const float bh = roih * (1.0f / (float)ROI);
    const float bw = roiw * (1.0f / (float)ROI);

    for (int p = 0; p < ROI; ++p) {
      const float pf = (float)p;
      int hs_ = (int)floorf(pf * bh) + sh;
      int he_ = (int)ceilf((pf + 1.0f) * bh) + sh;
      int ws_ = (int)floorf(pf * bw) + sw;
      int we_ = (int)ceilf((pf + 1.0f) * bw) + sw;
      P.hstart[p] = min(max(hs_, 0), H);
      P.hend[p]   = min(max(he_, 0), H);
      P.wstart[p] = min(max(ws_, 0), W);
      P.wend[p]   = min(max(we_, 0), W);
    }

    const int img = roi_idx[roi];
    P.base = fts[li] + (size_t)img * C_TOT * H * W;
    P.H = H;
    P.W = W;
    const int h0 = min(max(sh, 0), H);
    const int w0 = min(max(sw, 0), W);
    const int h1 = min(eh + 1, H);
    const int w1 = min(ew + 1, W);
    P.h0 = h0;
    P.w0 = w0;
    P.regH = min(max(h1 - h0, 0), TILE_H);
    P.regW = min(max(w1 - w0, 0), TILE_W);
  }
  __syncthreads();

  // ---- Stage the ROI footprint for CH_PER_BLK channels into LDS via the
  // ---- CDNA5 async global->LDS path (no VGPR round-trip, ASYNCcnt tracked).
  const int lane = threadIdx.x & 31;
  const int wv   = threadIdx.x >> 5;
  const int regH = P.regH;
  const int regW = P.regW;

  if (regH > 0 && regW > 0 && lane < regW) {
    const int HW = P.H * P.W;
    const int ntasks = CH_PER_BLK * regH;        // (channel,row) tasks
    for (int rt = wv; rt < ntasks; rt += 8) {
      const int cc = rt / regH;
      const int r  = rt - cc * regH;
      const float* g = P.base + (size_t)(cbase + cc) * HW
                              + (size_t)(P.h0 + r) * P.W + P.w0 + lane;
      const unsigned lo = (unsigned)(size_t)(const void*)&tile[cc][r][lane];
      async_ld_f32(g, lo);
    }
  }
  wait_async_zero();     // s_wait_asynccnt 0 (per-wave)
  __syncthreads();       // publish LDS tile to all waves

  // ---- 7x7 adaptive max-pool from LDS (each element reused by overlapping bins)
  const int h0 = P.h0, w0 = P.w0;
  for (int o = threadIdx.x; o < CH_PER_BLK * ROI * ROI; o += 256) {
    const int cc = o / (ROI * ROI);
    const int b  = o - cc * (ROI * ROI);
    const int ph = b / ROI;
    const int pw = b - ph * ROI;

    const int hs = P.hstart[ph], he = P.hend[ph];
    const int ws = P.wstart[pw], we = P.wend[pw];

    float res = 0.0f;
    if (he > hs && we > ws) {
      const int hl = min(he, hs + MAXB);   // MAX_BIN window cap
      const int wl = min(we, ws + MAXB);
      float m = -__builtin_inff();
      for (int h = hs; h < hl; ++h) {
        const float* row = &tile[cc][h - h0][0];
        for (int w = ws; w < wl; ++w) {
          m = fmaxf(m, row[w - w0]);
        }
      }
      res = m;
    }
    // Non-temporal store: keep the L2 full of feature-map lines, not output.
    __builtin_nontemporal_store(
        res, &out[(((size_t)roi * C_TOT + (cbase + cc)) * ROI + ph) * ROI + pw]);
  }
}

extern "C" void kernel_launch(void* const* d_in, const int* in_sizes, int n_in,
                              void* d_out, int out_size, void* d_ws, size_t ws_size,
                              hipStream_t stream) {
  (void)n_in; (void)out_size; (void)d_ws; (void)ws_size;
  const float* ft0 = (const float*)d_in[0];
  const float* ft1 = (const float*)d_in[1];
  const float* ft2 = (const float*)d_in[2];
  const float* ft3 = (const float*)d_in[3];
  const float* rois = (const float*)d_in[4];
  const int* roi_idx = (const int*)d_in[5];
  float* out = (float*)d_out;

  const int NB = in_sizes[4] / 4;                 // rois are (NB, 4)
  dim3 grid(NB, C_TOT / CH_PER_BLK);              // (256, 32)
  ROIPooler_kernel<<<grid, 256, 0, stream>>>(ft0, ft1, ft2, ft3, rois, roi_idx, out);
}